// TcSpace_31344671326417
// MI455X (gfx1250) — compile-verified
//
#include <hip/hip_runtime.h>
#include <hip/hip_bf16.h>
#include <math.h>

typedef __attribute__((ext_vector_type(2))) float v2f;
typedef __attribute__((ext_vector_type(8))) float v8f;

#define TT  8
#define GC  16
#define GGC 256
#define DC  64

// ---------------------------------------------------------------------------
// ws layout (floats): [0]=z_what_loss [1]=z_pres_loss [2]=sum(cos*pw) (pool)
// [3]=objects loss [4]=flow masked sq [5]=sum(zp) [6]=sum(flow)
// ---------------------------------------------------------------------------

__global__ void k_init(float* ws) {
    if (threadIdx.x < 7) ws[threadIdx.x] = 0.0f;
}

// ---- term 1: sum((zw[1:]-zw[:-1])^2), vectorized float4 --------------------
__global__ void k_zwhat(const float* __restrict__ zw, float* acc, int Fq, int n4) {
    __shared__ float s;
    if (threadIdx.x == 0) s = 0.0f;
    __syncthreads();
    int idx = blockIdx.x * blockDim.x + threadIdx.x;
    float p = 0.0f;
    if (idx < n4) {
        const float4* a = (const float4*)zw;
        float4 x = a[idx];
        float4 y = a[idx + Fq];   // Fq = frame stride in float4s
        float dx = y.x - x.x, dy = y.y - x.y, dz = y.z - x.z, dw = y.w - x.w;
        p = dx*dx + dy*dy + dz*dz + dw*dw;
    }
    atomicAdd(&s, p);
    __syncthreads();
    if (threadIdx.x == 0) atomicAdd(acc, s);
}

// ---- term 2: z_pres smoothness --------------------------------------------
__global__ void k_zpres(const float* __restrict__ zp, float* acc, int F2, int n) {
    __shared__ float s;
    if (threadIdx.x == 0) s = 0.0f;
    __syncthreads();
    int idx = blockIdx.x * blockDim.x + threadIdx.x;
    float p = 0.0f;
    if (idx < n) {
        float a = zp[idx];            // frame t
        float b = zp[idx + F2];       // frame t+1
        float c = zp[idx + 2 * F2];   // frame t+2
        float sim = 1.0f - (c - a) * (c - a);
        float del = (c - b) * (c - b) + (a - b) * (a - b);
        p = sim * del;
    }
    atomicAdd(&s, p);
    __syncthreads();
    if (threadIdx.x == 0) atomicAdd(acc, s);
}

// ---- term 3: pool-cosine. One WG per (t,b), t in 0..6. --------------------
__global__ void k_pool(const float* __restrict__ zw, const float* __restrict__ zp,
                       float* acc, int B) {
    extern __shared__ float sm[];
    float* w0 = sm;                 // 256*66
    float* w1 = w0 + 256 * 66;      // 256*66
    float* p0 = w1 + 256 * 66;      // 256
    float* p1 = p0 + 256;           // 256
    float* sa = p1 + 256;           // 1
    int tb = blockIdx.x, t = tb / B, b = tb % B;
    int tid = threadIdx.x;          // 256 threads
    const float* f0 = zw + (size_t)(t * B + b) * (GGC * DC);
    const float* f1 = zw + (size_t)((t + 1) * B + b) * (GGC * DC);
    p0[tid] = zp[(size_t)(t * B + b) * GGC + tid];
    p1[tid] = zp[(size_t)((t + 1) * B + b) * GGC + tid];
    if (tid == 0) sa[0] = 0.0f;
    __syncthreads();
    for (int u = tid; u < GGC * DC / 2; u += 256) {   // w = |zw| * zp
        int cell = u >> 5, ch = (u & 31) * 2;
        float q0 = p0[cell], q1 = p1[cell];
        w0[cell * 66 + ch]     = fabsf(f0[cell * DC + ch])     * q0;
        w0[cell * 66 + ch + 1] = fabsf(f0[cell * DC + ch + 1]) * q0;
        w1[cell * 66 + ch]     = fabsf(f1[cell * DC + ch])     * q1;
        w1[cell * 66 + ch + 1] = fabsf(f1[cell * DC + ch + 1]) * q1;
    }
    __syncthreads();
    int i = tid >> 4, j = tid & 15;
    int ia = i > 0 ? i - 1 : 0, ib = i < 15 ? i + 1 : 15;
    int ja = j > 0 ? j - 1 : 0, jb = j < 15 ? j + 1 : 15;
    float dot = 0.0f, na2 = 0.0f, nb2 = 0.0f;
    for (int d = 0; d < DC; ++d) {
        float mx = -INFINITY;                        // 3x3 clamped-border max
        for (int ii = ia; ii <= ib; ++ii)
            for (int jj = ja; jj <= jb; ++jj)
                mx = fmaxf(mx, w0[(ii * 16 + jj) * 66 + d]);
        float wv = w1[tid * 66 + d];
        dot += mx * wv; na2 += mx * mx; nb2 += wv * wv;
    }
    float na = fmaxf(sqrtf(na2), 1e-6f), nb = fmaxf(sqrtf(nb2), 1e-6f);
    float cosv = dot / (na * nb);
    float pw = 0.5f * (p0[tid] + p1[tid]);
    atomicAdd(sa, cosv * pw);
    __syncthreads();
    if (tid == 0) atomicAdd(acc, sa[0]);
}

// ---- term 4: objects loss, WMMA f32 16x16x4 Gram tiles. -------------------
// One WG per (t,b); 512 threads = 16 waves; wave i computes Gram(row i, rows
// i-1,i,i+1 of next frame) as 3 chained-WMMA 16x16x64 GEMMs.
__global__ void k_obj(const float* __restrict__ zw, const float* __restrict__ zp,
                      float* acc, int B) {
    extern __shared__ float sm[];
    float* P  = sm;                    // 256*68 prior frame (padded)
    float* Nx = P + 256 * 68;          // 256*68 next frame
    float* ST = Nx + 256 * 68;         // 16*3*256 Gram staging
    float* PN = ST + 16 * 3 * 256;     // 256 prior norms
    float* NN = PN + 256;              // 256 next norms
    float* ZP = NN + 256;              // 256 prior pres
    float* ZN = ZP + 256;              // 256 next pres
    float* SA = ZN + 256;              // 1
    int tb = blockIdx.x, t = tb / B, b = tb % B;
    int tid = threadIdx.x;             // 512 threads
    const float* f0 = zw + (size_t)(t * B + b) * (GGC * DC);
    const float* f1 = zw + (size_t)((t + 1) * B + b) * (GGC * DC);
    for (int u = tid; u < GGC * DC / 2; u += 512) {
        int cell = u >> 5, ch = (u & 31) * 2;
        P [cell * 68 + ch]     = f0[cell * DC + ch];
        P [cell * 68 + ch + 1] = f0[cell * DC + ch + 1];
        Nx[cell * 68 + ch]     = f1[cell * DC + ch];
        Nx[cell * 68 + ch + 1] = f1[cell * DC + ch + 1];
    }
    if (tid < GGC) ZP[tid] = zp[(size_t)(t * B + b) * GGC + tid];
    else           ZN[tid - GGC] = zp[(size_t)((t + 1) * B + b) * GGC + (tid - GGC)];
    if (tid == 0) SA[0] = 0.0f;
    __syncthreads();
    {   // norms: threads 0..255 -> prior, 256..511 -> next
        int c = tid & 255;
        const float* M = (tid < 256) ? P : Nx;
        float s = 0.0f;
        for (int d = 0; d < DC; ++d) { float v = M[c * 68 + d]; s += v * v; }
        float nv = fmaxf(sqrtf(s), 1e-8f);
        if (tid < 256) PN[c] = nv; else NN[c] = nv;
    }
    __syncthreads();
    // -------- WMMA phase: uniform control flow, EXEC all ones --------------
    int w = tid >> 5;                  // wave id == prior row i
    int lane = tid & 31;
    int hi = lane >> 4, col = lane & 15;
    int i = w;
    int ip0 = (i + 15) & 15, ip1 = i, ip2 = (i + 1) & 15;  // di = -1,0,+1 (wrap)
    // A lane layout (32b 16x4): lanes 0-15 M=0..15 K={k,k+1}; lanes 16-31 K={k+2,k+3}
    const float* Ab = P  + (i   * 16 + col) * 68 + 2 * hi;
    const float* B0 = Nx + (ip0 * 16 + col) * 68 + 2 * hi;  // B lane = N, same K striping
    const float* B1 = Nx + (ip1 * 16 + col) * 68 + 2 * hi;
    const float* B2 = Nx + (ip2 * 16 + col) * 68 + 2 * hi;
    v8f acc0 = {0,0,0,0,0,0,0,0};
    v8f acc1 = {0,0,0,0,0,0,0,0};
    v8f acc2 = {0,0,0,0,0,0,0,0};
    for (int k = 0; k < DC; k += 4) {
        v2f a  = *(const v2f*)(Ab + k);
        v2f b0 = *(const v2f*)(B0 + k);
        v2f b1 = *(const v2f*)(B1 + k);
        v2f b2 = *(const v2f*)(B2 + k);
        acc0 = __builtin_amdgcn_wmma_f32_16x16x4_f32(false, a, false, b0, (short)0, acc0, false, false);
        acc1 = __builtin_amdgcn_wmma_f32_16x16x4_f32(false, a, false, b1, (short)0, acc1, false, false);
        acc2 = __builtin_amdgcn_wmma_f32_16x16x4_f32(false, a, false, b2, (short)0, acc2, false, false);
    }
    // C/D layout: VGPR r holds M = r + 8*hi, N = col
    float* st = ST + w * 3 * 256;
#pragma unroll
    for (int r = 0; r < 8; ++r) {
        int M = r + 8 * hi;
        st[0 * 256 + M * 16 + col] = acc0[r];
        st[1 * 256 + M * 16 + col] = acc1[r];
        st[2 * 256 + M * 16 + col] = acc2[r];
    }
    __syncthreads();
    // -------- per-cell neighbor logic (16 lanes/wave, one cell each) -------
    float contrib = 0.0f;
    if (lane < 16) {
        int j = lane, c = i * 16 + j;
        float pn = PN[c], zpp = ZP[c];
        float ssum = 0.0f, smax = -INFINITY;
        bool has = false;
        int iprows[3] = {ip0, ip1, ip2};
        for (int a3 = 0; a3 < 3; ++a3) {
            int ip = iprows[a3];
            for (int dj = -1; dj <= 1; ++dj) {
                int jp = (j + dj) & 15;
                float dotv = st[a3 * 256 + j * 16 + jp];
                float s = dotv / (pn * NN[ip * 16 + jp]);
                if (ZN[ip * 16 + jp] > 0.5f) {
                    ssum += s; smax = fmaxf(smax, s); has = true;
                }
            }
        }
        if (zpp > 0.5f && has) contrib = -5.0f * smax + ssum;
    }
    atomicAdd(SA, contrib);
    __syncthreads();
    if (tid == 0) atomicAdd(acc, SA[0]);
}

// ---- term 5: flow loss pieces. One WG per image (T*B images). -------------
__global__ void k_flow(const float* __restrict__ zp, const float* __restrict__ fl,
                       float* ws) {
    __shared__ float img[GGC];
    __shared__ float s0, s1, s2;
    int im = blockIdx.x, tid = threadIdx.x;
    float zv = zp[(size_t)im * GGC + tid];
    img[tid] = zv;
    if (tid == 0) { s0 = 0.0f; s1 = 0.0f; s2 = 0.0f; }
    __syncthreads();
    int i = tid >> 4, j = tid & 15;
    int ia = i > 0 ? i - 1 : 0, ib = i < 15 ? i + 1 : 15;
    int ja = j > 0 ? j - 1 : 0, jb = j < 15 ? j + 1 : 15;
    float mx = -INFINITY;
    for (int ii = ia; ii <= ib; ++ii)
        for (int jj = ja; jj <= jb; ++jj)
            mx = fmaxf(mx, img[ii * 16 + jj]);
    float fv = fl[(size_t)im * GGC + tid];
    float term = (fv > 0.5f) ? (mx - fv) * (mx - fv) : 0.0f;
    atomicAdd(&s0, term);
    atomicAdd(&s1, zv);
    atomicAdd(&s2, fv);
    __syncthreads();
    if (tid == 0) {
        atomicAdd(&ws[4], s0);
        atomicAdd(&ws[5], s1);
        atomicAdd(&ws[6], s2);
    }
}

// ---- finalize -------------------------------------------------------------
__global__ void k_fin(const float* ws, const int* step, float* out) {
    if (threadIdx.x == 0 && blockIdx.x == 0) {
        float zwl = ws[0], zpl = ws[1], pool = ws[2], obj = ws[3];
        float fm = ws[4], zps = ws[5], fls = ws[6];
        float s = (float)step[0];
        float so = fminf(1.0f, s / 200000.0f);
        float sf = fmaxf(0.0f, 1.0f - s / 100000.0f);
        float flow_loss = fm + 100.0f * fmaxf(0.0f, zps - fls);
        out[0] = zwl * 1.0f + zpl * 1.0f + (-pool) * 1.0f
               + obj * so * 10.0f + flow_loss * 1.0f * sf;
    }
}

extern "C" void kernel_launch(void* const* d_in, const int* in_sizes, int n_in,
                              void* d_out, int out_size, void* d_ws, size_t ws_size,
                              hipStream_t stream) {
    const float* zw  = (const float*)d_in[0];
    const float* zp  = (const float*)d_in[1];
    const float* fl  = (const float*)d_in[2];
    const int*  step = (const int*)d_in[3];
    float* ws  = (float*)d_ws;
    float* out = (float*)d_out;

    int TB  = in_sizes[0] / (GGC * DC);   // T*B
    int B   = TB / TT;                    // batch
    int F   = B * GGC * DC;               // z_what frame stride (floats)
    int F2  = B * GGC;                    // z_pres frame stride (floats)

    k_init<<<1, 32, 0, stream>>>(ws);

    int n4 = (7 * F) / 4;
    k_zwhat<<<(n4 + 255) / 256, 256, 0, stream>>>(zw, ws + 0, F / 4, n4);

    int nP = 6 * F2;
    k_zpres<<<(nP + 255) / 256, 256, 0, stream>>>(zp, ws + 1, F2, nP);

    size_t poolLds = (size_t)(2 * 256 * 66 + 2 * 256 + 1) * sizeof(float);
    k_pool<<<7 * B, 256, poolLds, stream>>>(zw, zp, ws + 2, B);

    size_t objLds = (size_t)(2 * 256 * 68 + 16 * 3 * 256 + 4 * 256 + 1) * sizeof(float);
    k_obj<<<7 * B, 512, objLds, stream>>>(zw, zp, ws + 3, B);

    k_flow<<<TB, 256, 0, stream>>>(zp, fl, ws);

    k_fin<<<1, 1, 0, stream>>>(ws, step, out);
}